// TrainGRPOModulePallas_47854525612143
// MI455X (gfx1250) — compile-verified
//
#include <hip/hip_runtime.h>
#include <cstdint>

// Problem constants (from reference): B=4, S=1024, V=32000
#define B_    4
#define S_    1024
#define SM1   1023           // S-1
#define V_    32000
#define V4    8000           // V_/4 float4 per row (exact)
#define THR   512            // threads per row-block (16 waves of 32)
#define NWAVE (THR/32)
#define NT    16             // ceil(V4/THR) tiles per row
#define DEPTH 8              // async pipeline depth (LDS buffers, 64 KB)

// ---------------------------------------------------------------------------
// Kernel A: one block per (b,s) row. Single-pass online softmax statistics
// (m, Z = sum e^{x-m}, T = sum x*e^{x-m}) streamed through a DEPTH-deep
// async global->LDS pipeline (CDNA5 ASYNCcnt path), GVS addressing (SGPR row
// base + 32-bit lane offset), non-temporal since each byte is read once.
// Each thread copies and consumes only its own LDS slot -> no barriers in
// the streaming loop. Outputs per_token_logps[row] and token_entropy[row].
// ---------------------------------------------------------------------------
__global__ __launch_bounds__(THR) void grpo_row_kernel(
    const float* __restrict__ logits,
    const int*   __restrict__ input_ids,
    float* __restrict__ lp_out,     // -> d_out + 1, 4092 elements
    float* __restrict__ ent_out)    // -> workspace, 4092 elements
{
  __shared__ __align__(16) float4 buf[DEPTH][THR];
  __shared__ float rm[NWAVE], rz[NWAVE], rt[NWAVE];
  __shared__ float s_xc;

  const int tid = threadIdx.x;
  const int row = blockIdx.x;            // 0..4091
  const int b   = row / SM1;
  const int s   = row - b * SM1;
  const float* gbase = logits + (size_t)(b * S_ + s) * V_;

  if (tid == 0) {
    int chosen = input_ids[b * S_ + s + 1];
    s_xc = gbase[chosen];                // logit of the chosen token
  }

  // Low 32 bits of a generic shared pointer == LDS byte offset (aperture
  // encoding) == exactly what the async-load VDST operand wants.
  const uint32_t lds_base = (uint32_t)(uintptr_t)(void*)&buf[0][0];
  const uint64_t gbase64  = (uint64_t)(uintptr_t)gbase;   // uniform -> SGPR

  auto issue = [&](int t) {
    int idx = t * THR + tid;
    idx = (idx < V4) ? idx : (V4 - 1);   // clamp: every wave issues every tile
    uint32_t laddr = lds_base +
        (uint32_t)(((t & (DEPTH - 1)) * THR + tid) * sizeof(float4));
    uint32_t goff = (uint32_t)idx * (uint32_t)sizeof(float4);
    asm volatile("global_load_async_to_lds_b128 %0, %1, %2 th:TH_LOAD_NT"
                 :: "v"(laddr), "v"(goff), "s"(gbase64) : "memory");
  };

  float m = -__builtin_inff(), Z = 0.f, Tq = 0.f;

  auto process = [&](int t) {
    int idx = t * THR + tid;
    if (idx < V4) {
      float4 v = buf[t & (DEPTH - 1)][tid];   // ds_load_b128 (own slot)
      float mv = fmaxf(fmaxf(v.x, v.y), fmaxf(v.z, v.w));
      if (mv > m) {                            // rare after warmup
        float c = __expf(m - mv);
        Z *= c; Tq *= c; m = mv;
      }
      float e0 = __expf(v.x - m), e1 = __expf(v.y - m);
      float e2 = __expf(v.z - m), e3 = __expf(v.w - m);
      Z  += (e0 + e1) + (e2 + e3);
      Tq += (v.x * e0 + v.y * e1) + (v.z * e2 + v.w * e3);
    }
  };

  // Prologue: fill the pipeline (tiles 0..DEPTH-1).
  issue(0); issue(1); issue(2); issue(3);
  issue(4); issue(5); issue(6); issue(7);
  // Steady state: keep DEPTH tiles in flight per wave.
  for (int t = 0; t < NT - DEPTH; ++t) {
    asm volatile("s_wait_asynccnt 7" ::: "memory");  // tile t complete
    process(t);
    issue(t + DEPTH);
  }
  // Drain: outstanding tiles NT-DEPTH .. NT-1.
  asm volatile("s_wait_asynccnt 7" ::: "memory"); process(NT - 8);
  asm volatile("s_wait_asynccnt 6" ::: "memory"); process(NT - 7);
  asm volatile("s_wait_asynccnt 5" ::: "memory"); process(NT - 6);
  asm volatile("s_wait_asynccnt 4" ::: "memory"); process(NT - 5);
  asm volatile("s_wait_asynccnt 3" ::: "memory"); process(NT - 4);
  asm volatile("s_wait_asynccnt 2" ::: "memory"); process(NT - 3);
  asm volatile("s_wait_asynccnt 1" ::: "memory"); process(NT - 2);
  asm volatile("s_wait_asynccnt 0" ::: "memory"); process(NT - 1);

  // Wave32 reduction of (m, Z, T).
  for (int o = 16; o > 0; o >>= 1) {
    float m2 = __shfl_xor(m,  o, 32);
    float z2 = __shfl_xor(Z,  o, 32);
    float t2 = __shfl_xor(Tq, o, 32);
    float mm = fmaxf(m, m2);
    float c1 = __expf(m - mm), c2 = __expf(m2 - mm);
    Z = Z * c1 + z2 * c2;
    Tq = Tq * c1 + t2 * c2;
    m = mm;
  }
  const int wid = tid >> 5, lane = tid & 31;
  if (lane == 0) { rm[wid] = m; rz[wid] = Z; rt[wid] = Tq; }
  __syncthreads();

  if (wid == 0) {
    float mA = (lane < NWAVE) ? rm[lane] : -__builtin_inff();
    float zA = (lane < NWAVE) ? rz[lane] : 0.f;
    float tA = (lane < NWAVE) ? rt[lane] : 0.f;
    for (int o = 16; o > 0; o >>= 1) {
      float m2 = __shfl_xor(mA, o, 32);
      float z2 = __shfl_xor(zA, o, 32);
      float t2 = __shfl_xor(tA, o, 32);
      float mm = fmaxf(mA, m2);
      float c1 = __expf(mA - mm), c2 = __expf(m2 - mm);
      zA = zA * c1 + z2 * c2;
      tA = tA * c1 + t2 * c2;
      mA = mm;
    }
    if (lane == 0) {
      float lse = mA + logf(zA);          // log-sum-exp of the row
      lp_out[row]  = s_xc - lse;          // per-token logp (TEMP == 1)
      ent_out[row] = lse - tA / zA;       // -sum p*log p
    }
  }
}

// ---------------------------------------------------------------------------
// Kernel B: per-sample entropy averages. Needs cumsum(valid) for the
// 4 <= cum <= 100 truncation window -> block-wide Hillis-Steele scan.
// ---------------------------------------------------------------------------
__global__ __launch_bounds__(1024) void grpo_finalize_kernel(
    const int*   __restrict__ labels,
    const float* __restrict__ ent,
    float* __restrict__ out)             // full d_out
{
  const int b = blockIdx.x, tid = threadIdx.x;
  __shared__ int   sc[1024];
  __shared__ float red[1024];

  int valid = 0; float E = 0.f;
  if (tid < SM1) {
    valid = labels[b * S_ + tid + 1];    // labels in {0,1}
    E     = ent[b * SM1 + tid];
  }
  sc[tid] = valid;
  __syncthreads();
  for (int o = 1; o < 1024; o <<= 1) {   // inclusive scan
    int v = (tid >= o) ? sc[tid - o] : 0;
    __syncthreads();
    sc[tid] += v;
    __syncthreads();
  }
  const int cum = sc[tid];
  const bool tr = valid && (cum >= 4) && (cum <= 100);

  float vals[4] = { valid ? E : 0.f, (float)valid, tr ? E : 0.f, tr ? 1.f : 0.f };
  float sums[4];
  for (int k = 0; k < 4; ++k) {
    red[tid] = vals[k];
    __syncthreads();
    for (int st = 512; st > 0; st >>= 1) {
      if (tid < st) red[tid] += red[tid + st];
      __syncthreads();
    }
    sums[k] = red[0];
    __syncthreads();
  }
  if (tid == 0) {
    out[1 + B_ * SM1 + b]      = sums[0] / sums[1];  // avg_entropy_per_sample
    out[1 + B_ * SM1 + B_ + b] = sums[2] / sums[3];  // avg_entropy_truncated
  }
}

// ---------------------------------------------------------------------------
// Kernel C: the loss. ratio = exp(lp - stopgrad(lp)) == 1 identically, so
// per_token_loss = -adv[b]; loss = -(sum adv_b*mask)/(sum mask).
// ---------------------------------------------------------------------------
__global__ __launch_bounds__(256) void grpo_loss_kernel(
    const int*   __restrict__ labels,
    const float* __restrict__ adv,
    float* __restrict__ out)             // out[0] = loss
{
  __shared__ float sn[256], sd[256];
  const int tid = threadIdx.x;
  float num = 0.f, den = 0.f;
  for (int i = tid; i < B_ * SM1; i += 256) {
    int b = i / SM1, s = i - b * SM1;
    float mk = (float)labels[b * S_ + s + 1];
    num += mk * adv[b];
    den += mk;
  }
  sn[tid] = num; sd[tid] = den;
  __syncthreads();
  for (int st = 128; st > 0; st >>= 1) {
    if (tid < st) { sn[tid] += sn[tid + st]; sd[tid] += sd[tid + st]; }
    __syncthreads();
  }
  if (tid == 0) out[0] = -sn[0] / sd[0];
}

// ---------------------------------------------------------------------------
extern "C" void kernel_launch(void* const* d_in, const int* in_sizes, int n_in,
                              void* d_out, int out_size, void* d_ws, size_t ws_size,
                              hipStream_t stream) {
  (void)in_sizes; (void)n_in; (void)out_size; (void)ws_size;
  const float* logits    = (const float*)d_in[0];
  const int*   input_ids = (const int*)d_in[1];
  const int*   labels    = (const int*)d_in[2];
  const float* adv       = (const float*)d_in[3];
  float* out = (float*)d_out;            // [loss | lp(4092) | avgE(4) | avgEtr(4)]
  float* ent = (float*)d_ws;             // 4092 floats of scratch

  grpo_row_kernel<<<B_ * SM1, THR, 0, stream>>>(logits, input_ids, out + 1, ent);
  grpo_finalize_kernel<<<B_, 1024, 0, stream>>>(labels, ent, out);
  grpo_loss_kernel<<<1, 256, 0, stream>>>(labels, adv, out);
}